// EfficientByteMUL_7945689497962
// MI455X (gfx1250) — compile-verified
//
#include <hip/hip_runtime.h>

typedef __attribute__((ext_vector_type(2))) float v2f;
typedef __attribute__((ext_vector_type(8))) float v8f;

#define BD_DIM 128
#define HID    256
#define GE_DIM 64

// out = copy of x with +2.0*mask one-hots at 80+r_lo / 96+r_hi, where
// res byte = round( relu-dot MLP(byte_a, byte_b) ) & 255.
__global__ __launch_bounds__(256) void ebm_fused_kernel(
    const float* __restrict__ x,
    const float* __restrict__ W1, const float* __restrict__ b1,
    const float* __restrict__ W2, const float* __restrict__ b2,
    float* __restrict__ out, long T)
{
    const int  lane = threadIdx.x & 31;
    const long gw   = (long)blockIdx.x * 8 + (threadIdx.x >> 5);
    const long tb   = gw * 16;
    if (tb >= T) return;

    const float b2r = b2[40];

    if (tb + 16 <= T) {
        // ---------- fast path: 16 full tokens per wave ----------
        // mask preload: lane m (<16) owns token tb+m
        int maskbit = 0;
        if (lane < 16) {
            const float* p = x + (tb + lane) * BD_DIM;
            maskbit = (p[0] >= 0.5f && p[1] >= 0.5f) ? 1 : 0;
        }

        // decode: 4 argmax groups (floats 16..79), 8 lanes per group, 2 vals/lane
        int av = 0, bv = 0;                 // lane m keeps token m's bytes
        const unsigned gpos = 2u * (unsigned)(lane & 7);
        for (int m = 0; m < 16; ++m) {
            const float* p = x + (tb + m) * BD_DIM + 16;
            float x0 = p[2 * lane];
            float x1 = p[2 * lane + 1];
            unsigned k0 = (__float_as_uint(x0) & 0xFFFFFFF0u) | gpos;
            unsigned k1 = (__float_as_uint(x1) & 0xFFFFFFF0u) | (gpos + 1u);
            unsigned k  = k0 > k1 ? k0 : k1;
            unsigned o;
            o = (unsigned)__shfl_xor((int)k, 1); k = k > o ? k : o;
            o = (unsigned)__shfl_xor((int)k, 2); k = k > o ? k : o;
            o = (unsigned)__shfl_xor((int)k, 4); k = k > o ? k : o;
            int nib = (int)(k & 15u);
            int n0 = __shfl(nib, 0),  n1 = __shfl(nib, 8);
            int n2 = __shfl(nib, 16), n3 = __shfl(nib, 24);
            if (lane == m) { av = n0 | (n1 << 4); bv = n2 | (n3 << 4); }
        }

        // layer 1 via V_WMMA_F32_16X16X4_F32, fused relu + W2[:,40] dot
        // A (16x4): lanes 0-15 row=lane {K0=byte_a,K1=byte_b}; lanes 16-31 {K2=1,K3=0}
        v2f a;
        a.x = (lane < 16) ? (float)av : 1.0f;
        a.y = (lane < 16) ? (float)bv : 0.0f;

        float acc[8] = {0.f,0.f,0.f,0.f,0.f,0.f,0.f,0.f};
        const int col0 = lane & 15;
        for (int t = 0; t < 16; ++t) {
            const int c = t * 16 + col0;
            // B (4x16) tile: K0=W1[0,c], K1=W1[1,c], K2=W1[29,c]+b1[c], K3=0
            v2f b;
            if (lane < 16) { b.x = W1[c];            b.y = W1[HID + c]; }
            else           { b.x = W1[29*HID + c] + b1[c]; b.y = 0.0f;  }
            v8f cz = {};
            v8f d = __builtin_amdgcn_wmma_f32_16x16x4_f32(
                        false, a, false, b, (short)0, cz, false, false);
            const float w2v = W2[(size_t)c * GE_DIM + 40];
#pragma unroll
            for (int r = 0; r < 8; ++r) {
                float h = d[r] > 0.0f ? d[r] : 0.0f;
                acc[r] += h * w2v;
            }
        }
        // reduce each 16-lane half (rows 0-7 in lanes<16, rows 8-15 in lanes>=16)
#pragma unroll
        for (int r = 0; r < 8; ++r) {
            acc[r] += __shfl_xor(acc[r], 1);
            acc[r] += __shfl_xor(acc[r], 2);
            acc[r] += __shfl_xor(acc[r], 4);
            acc[r] += __shfl_xor(acc[r], 8);
        }
        // redistribute: lane m (<16) gets token m's scalar
        float tot = 0.0f;
#pragma unroll
        for (int r = 0; r < 8; ++r) {
            float vlo = __shfl(acc[r], 0);
            float vhi = __shfl(acc[r], 16);
            if (lane == r)     tot = vlo;
            if (lane == r + 8) tot = vhi;
        }
        int pk = 0;
        if (lane < 16) {
            int res = ((int)roundf(tot + b2r)) & 255;
            pk = (maskbit << 8) | res;
        }

        // apply: stream-copy 16 tokens (float4/lane, coalesced) with sparse adds
        const float4* in4  = (const float4*)x;
        float4*       out4 = (float4*)out;
        const int base = lane * 4;
        for (int i = 0; i < 16; ++i) {
            int p = __shfl(pk, i);
            long idx = (tb + i) * 32 + lane;
            float4 v = in4[idx];
            if (p & 0x100) {
                int f0 = 80 + (p & 15);
                int f1 = 96 + ((p >> 4) & 15);
                v.x += (base + 0 == f0 || base + 0 == f1) ? 2.0f : 0.0f;
                v.y += (base + 1 == f0 || base + 1 == f1) ? 2.0f : 0.0f;
                v.z += (base + 2 == f0 || base + 2 == f1) ? 2.0f : 0.0f;
                v.w += (base + 3 == f0 || base + 3 == f1) ? 2.0f : 0.0f;
            }
            out4[idx] = v;
        }
    } else {
        // ---------- scalar tail (T not a multiple of 16; dead for 8x8192) ----------
        for (long t = tb + lane; t < T; t += 32) {
            const float* p = x + t * BD_DIM;
            float*       o = out + t * BD_DIM;
            bool msk = (p[0] >= 0.5f) && (p[1] >= 0.5f);
            int nib[4];
            for (int g = 0; g < 4; ++g) {
                const float* q = p + 16 + g * 16;
                int best = 0; float bmax = q[0];
                for (int i = 1; i < 16; ++i)
                    if (q[i] > bmax) { bmax = q[i]; best = i; }
                nib[g] = best;
            }
            float ba = (float)(nib[0] | (nib[1] << 4));
            float bb = (float)(nib[2] | (nib[3] << 4));
            float tot = b2r;
            for (int k = 0; k < HID; ++k) {
                float h = ba * W1[k] + bb * W1[HID + k] + W1[29*HID + k] + b1[k];
                h = h > 0.0f ? h : 0.0f;
                tot += h * W2[(size_t)k * GE_DIM + 40];
            }
            int res = ((int)roundf(tot)) & 255;
            int f0 = 80 + (res & 15);
            int f1 = 96 + ((res >> 4) & 15);
            for (int i = 0; i < BD_DIM; ++i) {
                float v = p[i];
                if (msk && (i == f0 || i == f1)) v += 2.0f;
                o[i] = v;
            }
        }
    }
}

extern "C" void kernel_launch(void* const* d_in, const int* in_sizes, int n_in,
                              void* d_out, int out_size, void* d_ws, size_t ws_size,
                              hipStream_t stream) {
    const float* x  = (const float*)d_in[0];
    const float* W1 = (const float*)d_in[1];
    const float* b1 = (const float*)d_in[2];
    const float* W2 = (const float*)d_in[3];
    const float* b2 = (const float*)d_in[4];
    float* out = (float*)d_out;

    long T = (long)in_sizes[0] / BD_DIM;          // 65536 tokens
    long waves  = (T + 15) / 16;                   // 16 tokens per wave32
    long blocks = (waves + 7) / 8;                 // 8 waves (256 threads) per block
    if (blocks < 1) blocks = 1;
    ebm_fused_kernel<<<(int)blocks, 256, 0, stream>>>(x, W1, b1, W2, b2, out, T);
}